// GCN_52029233824482
// MI455X (gfx1250) — compile-verified
//
#include <hip/hip_runtime.h>
#include <hip/hip_bf16.h>

typedef __attribute__((ext_vector_type(2))) float v2f;
typedef __attribute__((ext_vector_type(8))) float v8f;

#define GCN_IN_C  128
#define GCN_HID_C 64

// -----------------------------------------------------------------------------
// GEMM: H[r, 0:64] = act(A[r, 0:K]) @ W[K, 64]
//   act(v) = bias ? relu(v + bias[c]) : v   (fuses previous layer's bias+ReLU)
// One block = 256 threads = 8 waves; each wave owns 16 rows x 64 cols.
// A tile staged in LDS in 64-wide K chunks (padded, conflict-free).
// Matrix core: V_WMMA_F32_16X16X4_F32 (exact f32, matches f32 reference).
// -----------------------------------------------------------------------------
template <int K>
__global__ __launch_bounds__(256) void gcn_gemm_wmma(
    const float* __restrict__ A, const float* __restrict__ W,
    const float* __restrict__ bias, float* __restrict__ H, int nRows)
{
    constexpr int ROWS = 128;       // rows per block (8 waves * 16)
    constexpr int KC   = 64;        // K chunk staged in LDS
    constexpr int LDK  = KC + 4;    // pad: 4m+k bank pattern, conflict-free
    __shared__ float As[ROWS * LDK];   // 128*68*4 = 34816 B

    const int row0 = blockIdx.x * ROWS;
    const int wave = threadIdx.x >> 5;
    const int lane = threadIdx.x & 31;
    const int m    = lane & 15;          // row-in-tile (A) / col (B,C)
    const int kk   = (lane >> 4) << 1;   // 0 or 2: which K pair this half-wave holds

    v8f acc0 = {}, acc1 = {}, acc2 = {}, acc3 = {};

    for (int k0 = 0; k0 < K; k0 += KC) {
        __syncthreads();
        // ---- stage ROWS x KC chunk of A into LDS (coalesced float4) ----
        for (int i = threadIdx.x; i < ROWS * (KC / 4); i += 256) {
            const int r  = i / (KC / 4);
            const int c  = (i % (KC / 4)) * 4;
            const int gr = row0 + r;
            const int gc = k0 + c;
            float4 v = make_float4(0.f, 0.f, 0.f, 0.f);
            if (gr < nRows)
                v = *(const float4*)(A + (size_t)gr * K + gc);
            if (bias) {  // fused bias + ReLU from previous layer
                v.x = fmaxf(v.x + bias[gc + 0], 0.f);
                v.y = fmaxf(v.y + bias[gc + 1], 0.f);
                v.z = fmaxf(v.z + bias[gc + 2], 0.f);
                v.w = fmaxf(v.w + bias[gc + 3], 0.f);
            }
            *(float4*)(&As[r * LDK + c]) = v;
        }
        __syncthreads();

        const float* arow = &As[(wave * 16 + m) * LDK];
        #pragma unroll
        for (int k = 0; k < KC; k += 4) {
            // A fragment 16x4: lanes 0-15 hold K=k..k+1, lanes 16-31 K=k+2..k+3
            v2f a;
            a.x = arow[k + kk + 0];
            a.y = arow[k + kk + 1];
            // B fragments 4x16: v0 = row (k+kk), v1 = row (k+kk+1), col n0+m
            const float* b0p = W + (size_t)(k0 + k + kk + 0) * 64 + m;
            const float* b1p = W + (size_t)(k0 + k + kk + 1) * 64 + m;
            v2f b;
            b.x = b0p[0];  b.y = b1p[0];
            acc0 = __builtin_amdgcn_wmma_f32_16x16x4_f32(false, a, false, b, (short)0, acc0, false, false);
            b.x = b0p[16]; b.y = b1p[16];
            acc1 = __builtin_amdgcn_wmma_f32_16x16x4_f32(false, a, false, b, (short)0, acc1, false, false);
            b.x = b0p[32]; b.y = b1p[32];
            acc2 = __builtin_amdgcn_wmma_f32_16x16x4_f32(false, a, false, b, (short)0, acc2, false, false);
            b.x = b0p[48]; b.y = b1p[48];
            acc3 = __builtin_amdgcn_wmma_f32_16x16x4_f32(false, a, false, b, (short)0, acc3, false, false);
        }
    }

    // C/D layout: VGPR i -> M = i + (lane<16 ? 0 : 8), N = lane&15
    const int rbase = row0 + wave * 16 + (lane >> 4) * 8;
    #pragma unroll
    for (int i = 0; i < 8; ++i) {
        const int r = rbase + i;
        if (r < nRows) {
            float* out = H + (size_t)r * 64 + m;
            out[0]  = acc0[i];
            out[16] = acc1[i];
            out[32] = acc2[i];
            out[48] = acc3[i];
        }
    }
}

// -----------------------------------------------------------------------------
// Scatter-add: agg[dst[e]] += h[src[e]].  16 threads per edge, float4 per
// thread, 4 non-returning global_atomic_add_f32 each. agg (25.6MB) is
// L2-resident on MI455X (192MB L2).
// -----------------------------------------------------------------------------
__global__ __launch_bounds__(256) void gcn_scatter_add(
    const float* __restrict__ h, const long long* __restrict__ src,
    const long long* __restrict__ dst, float* __restrict__ agg, int nEdges)
{
    const int t = blockIdx.x * 256 + threadIdx.x;
    if (t >= nEdges * 16) return;
    const int e = t >> 4;
    const int c = (t & 15) << 2;
    const long long s = src[e];
    const long long d = dst[e];
    const float4 v = *(const float4*)(h + (size_t)s * 64 + c);
    float* p = agg + (size_t)d * 64 + c;
    unsafeAtomicAdd(p + 0, v.x);
    unsafeAtomicAdd(p + 1, v.y);
    unsafeAtomicAdd(p + 2, v.z);
    unsafeAtomicAdd(p + 3, v.w);
}

__global__ __launch_bounds__(256) void gcn_zero_f4(float4* __restrict__ p, int n4)
{
    const int t = blockIdx.x * 256 + threadIdx.x;
    if (t < n4) p[t] = make_float4(0.f, 0.f, 0.f, 0.f);
}

// Final layer: out = relu(agg + b3)
__global__ __launch_bounds__(256) void gcn_bias_relu(
    const float* __restrict__ agg, const float* __restrict__ b,
    float* __restrict__ out, int nNodes)
{
    const int t = blockIdx.x * 256 + threadIdx.x;
    if (t >= nNodes * 16) return;
    const int c = (t & 15) << 2;
    float4 v = ((const float4*)agg)[t];
    v.x = fmaxf(v.x + b[c + 0], 0.f);
    v.y = fmaxf(v.y + b[c + 1], 0.f);
    v.z = fmaxf(v.z + b[c + 2], 0.f);
    v.w = fmaxf(v.w + b[c + 3], 0.f);
    ((float4*)out)[t] = v;
}

extern "C" void kernel_launch(void* const* d_in, const int* in_sizes, int n_in,
                              void* d_out, int out_size, void* d_ws, size_t ws_size,
                              hipStream_t stream)
{
    const float*     x   = (const float*)d_in[0];
    const long long* src = (const long long*)d_in[1];   // jnp.int64
    const long long* dst = (const long long*)d_in[2];
    const float*     W1  = (const float*)d_in[3];
    const float*     b1  = (const float*)d_in[4];
    const float*     W2  = (const float*)d_in[5];
    const float*     b2  = (const float*)d_in[6];
    const float*     W3  = (const float*)d_in[7];
    const float*     b3  = (const float*)d_in[8];

    const int nNodes = in_sizes[0] / GCN_IN_C;
    const int nEdges = in_sizes[1];

    float* h   = (float*)d_ws;                                          // [N,64]
    float* agg = (float*)((char*)d_ws + (size_t)nNodes * GCN_HID_C * sizeof(float));
    float* out = (float*)d_out;

    const int n4       = nNodes * GCN_HID_C / 4;
    const int gemmGrid = (nNodes + 127) / 128;
    const int scatGrid = (nEdges * 16 + 255) / 256;
    const int eltGrid  = (n4 + 255) / 256;
    const dim3 blk(256);

    // Layer 1: h = x @ W1 ; agg = scatter_add(h[src], dst)
    gcn_gemm_wmma<GCN_IN_C><<<gemmGrid, blk, 0, stream>>>(x, W1, nullptr, h, nNodes);
    gcn_zero_f4<<<eltGrid, blk, 0, stream>>>((float4*)agg, n4);
    gcn_scatter_add<<<scatGrid, blk, 0, stream>>>(h, src, dst, agg, nEdges);

    // Layer 2: h = relu(agg + b1) @ W2 (bias+ReLU fused into GEMM A-load)
    gcn_gemm_wmma<GCN_HID_C><<<gemmGrid, blk, 0, stream>>>(agg, W2, b1, h, nNodes);
    gcn_zero_f4<<<eltGrid, blk, 0, stream>>>((float4*)agg, n4);
    gcn_scatter_add<<<scatGrid, blk, 0, stream>>>(h, src, dst, agg, nEdges);

    // Layer 3: h = relu(agg + b2) @ W3
    gcn_gemm_wmma<GCN_HID_C><<<gemmGrid, blk, 0, stream>>>(agg, W3, b2, h, nNodes);
    gcn_zero_f4<<<eltGrid, blk, 0, stream>>>((float4*)agg, n4);
    gcn_scatter_add<<<scatGrid, blk, 0, stream>>>(h, src, dst, agg, nEdges);

    // out = relu(agg + b3)
    gcn_bias_relu<<<eltGrid, blk, 0, stream>>>(agg, b3, out, nNodes);
}